// Generator_56513179681265
// MI455X (gfx1250) — compile-verified
//
#include <hip/hip_runtime.h>
#include <math.h>

// CDNA5 / gfx1250, wave32.
typedef __attribute__((ext_vector_type(2))) float v2f;
typedef __attribute__((ext_vector_type(8))) float v8f;

#define WAVE 32
#define MT 2   // M-tiles (16 rows each) per wave
#define NT 4   // N-tiles (16 cols each) per wave

__device__ __forceinline__ int imin(int a, int b) { return a < b ? a : b; }

#define WMMA_F32(a, b, c) \
  __builtin_amdgcn_wmma_f32_16x16x4_f32(false, (a), false, (b), (short)0, (c), false, false)

// ---------------------------------------------------------------------------
// Fast WMMA GEMM:  C[m,n] = sum_k A[m,k] * W[n,k] (+bias[n])  == A @ W^T
// Requirements (host-dispatched): M%16==0, N%16==0, K%4==0, tilesM>=MT,
// tilesN>=NT.  One wave computes a 32x64 C block (2x4 tiles, 8 accumulators).
//
// Edge handling: group start tiles are SHIFTED down (min(g*T, tiles-T)) so all
// loads stay in-bounds with uniform 16*K inter-tile stride; overlapping tiles
// are computed but only stored by the owning group (store guard outside loop).
//
// Per K4 step: 2 A + 4 B global_load_b64 (named scalars -> distinct VGPRs,
// batched issue, single wait) feed 8 independent v_wmma_f32_16x16x4_f32.
// KC>0 bakes K in as a constant so tile strides become instruction immediates.
// ---------------------------------------------------------------------------
template <int KC>
__global__ __launch_bounds__(128)
void gemm_fast(const float* __restrict__ A, const float* __restrict__ W,
               const float* __restrict__ bias, float* __restrict__ C,
               int M, int N, int Krt) {
  const int K = KC ? KC : Krt;
  const int tilesM  = M >> 4;
  const int tilesN  = N >> 4;
  const int mGroups = (tilesM + MT - 1) / MT;
  const int nGroups = (tilesN + NT - 1) / NT;
  const int wid = blockIdx.x * (blockDim.x / WAVE) + ((int)threadIdx.x / WAVE);
  if (wid >= mGroups * nGroups) return;
  const int mg = wid / nGroups;
  const int ng = wid % nGroups;
  const int mStart = imin(mg * MT, tilesM - MT);   // shifted, always in-bounds
  const int nStart = imin(ng * NT, tilesN - NT);

  const int lane = (int)threadIdx.x & (WAVE - 1);
  const int half = lane >> 4;          // 0: lanes 0-15, 1: lanes 16-31
  const int l16  = lane & 15;
  const int kh   = half * 2;           // lane K sub-offset {0,2}

  const int row0 = mStart * 16 + l16;
  const int col0 = nStart * 16 + l16;
  const float* ap = A + (size_t)row0 * K + kh;
  const float* wp = W + (size_t)col0 * K + kh;
  const size_t ts = (size_t)16 * K;    // inter-tile element stride (immediate if KC)

  v8f c00 = {}, c01 = {}, c02 = {}, c03 = {};
  v8f c10 = {}, c11 = {}, c12 = {}, c13 = {};

  for (int k0 = 0; k0 < K; k0 += 4) {
    const v2f a0 = *(const v2f*)(ap);
    const v2f a1 = *(const v2f*)(ap + ts);
    const v2f b0 = *(const v2f*)(wp);
    const v2f b1 = *(const v2f*)(wp + ts);
    const v2f b2 = *(const v2f*)(wp + 2 * ts);
    const v2f b3 = *(const v2f*)(wp + 3 * ts);
    ap += 4; wp += 4;
    c00 = WMMA_F32(a0, b0, c00);
    c01 = WMMA_F32(a0, b1, c01);
    c02 = WMMA_F32(a0, b2, c02);
    c03 = WMMA_F32(a0, b3, c03);
    c10 = WMMA_F32(a1, b0, c10);
    c11 = WMMA_F32(a1, b1, c11);
    c12 = WMMA_F32(a1, b2, c12);
    c13 = WMMA_F32(a1, b3, c13);
  }

  // store owned tiles (shifted duplicates skipped; edge-free since M,N %16==0)
  const v8f* accs[MT][NT] = {{&c00, &c01, &c02, &c03}, {&c10, &c11, &c12, &c13}};
#pragma unroll
  for (int mi = 0; mi < MT; ++mi) {
    if (mStart + mi < mg * MT) continue;        // owned by an earlier group
#pragma unroll
    for (int nj = 0; nj < NT; ++nj) {
      if (nStart + nj < ng * NT) continue;
      const int n = col0 + nj * 16;
      const float bv = bias ? bias[n] : 0.f;
      const int mb = (mStart + mi) * 16 + half * 8;
      const v8f& ac = *accs[mi][nj];
#pragma unroll
      for (int r = 0; r < 8; ++r)
        C[(size_t)(mb + r) * N + n] = ac[r] + bv;
    }
  }
}

// ---------------------------------------------------------------------------
// Small/edge WMMA GEMM: one 16x16 tile per wave, clamped indices (branch-free
// loop). Used for tiny shapes (q: M=8, gates: N=3).
// ---------------------------------------------------------------------------
__global__ __launch_bounds__(128)
void gemm_small(const float* __restrict__ A, const float* __restrict__ W,
                const float* __restrict__ bias, float* __restrict__ C,
                int M, int N, int K) {
  const int tilesN = (N + 15) >> 4;
  const int tilesM = (M + 15) >> 4;
  const int wid = blockIdx.x * (blockDim.x / WAVE) + ((int)threadIdx.x / WAVE);
  if (wid >= tilesM * tilesN) return;
  const int tm = wid / tilesN;
  const int tn = wid % tilesN;
  const int lane = (int)threadIdx.x & (WAVE - 1);
  const int half = lane >> 4;
  const int l16  = lane & 15;
  const int kh   = half * 2;

  const int row = tm * 16 + l16;
  const int col = tn * 16 + l16;
  const int rowc = imin(row, M - 1);   // clamped garbage only feeds discarded C rows
  const int colc = imin(col, N - 1);
  const float* ap = A + (size_t)rowc * K + kh;
  const float* wp = W + (size_t)colc * K + kh;

  v8f acc = {};
  for (int k0 = 0; k0 < K; k0 += 4) {
    const v2f a = *(const v2f*)ap; ap += 4;
    const v2f b = *(const v2f*)wp; wp += 4;
    acc = WMMA_F32(a, b, acc);
  }
  if (col < N) {
    const float bv = bias ? bias[col] : 0.f;
    const int mbase = tm * 16 + half * 8;
#pragma unroll
    for (int r = 0; r < 8; ++r) {
      const int m = mbase + r;
      if (m < M) C[(size_t)m * N + col] = acc[r] + bv;
    }
  }
}

// ---------------------------------------------------------------------------
// Per-(s,b) masked attention: att[l] = <query, enc[b,l,:]> (masked), softmax
// over l, prob -> prob_out, ctx[h] = sum_l prob[l]*enc[b,l,h] -> ctx_out.
// Dynamic LDS: H floats (query) + L floats (scores).
// ---------------------------------------------------------------------------
__global__ __launch_bounds__(256)
void attn_softmax_ctx(const float* __restrict__ query, int qs_s, int qs_b,
                      const float* __restrict__ enc, const int* __restrict__ lens,
                      float* __restrict__ prob_out, float* __restrict__ ctx_out,
                      int B, int L, int H) {
  extern __shared__ float sm[];
  float* qsm = sm;          // [H]
  float* att = sm + H;      // [L]
  __shared__ float red[256];
  const int sb = blockIdx.x;
  const int s  = sb / B;
  const int b  = sb % B;
  const int tid = threadIdx.x;
  const float* q  = query + (size_t)s * qs_s + (size_t)b * qs_b;
  const float* eb = enc + (size_t)b * L * H;
  const int len = lens[b];

  for (int h = tid; h < H; h += 256) qsm[h] = q[h];
  __syncthreads();
  for (int l = tid; l < L; l += 256) {
    float d = 0.f;
    const float* e = eb + (size_t)l * H;
    for (int h = 0; h < H; ++h) d += qsm[h] * e[h];
    att[l] = (l < len) ? d : -1e30f;
  }
  __syncthreads();
  float m = -1e30f;
  for (int l = tid; l < L; l += 256) m = fmaxf(m, att[l]);
  red[tid] = m; __syncthreads();
  for (int st = 128; st > 0; st >>= 1) {
    if (tid < st) red[tid] = fmaxf(red[tid], red[tid + st]);
    __syncthreads();
  }
  m = red[0]; __syncthreads();
  float ssum = 0.f;
  for (int l = tid; l < L; l += 256) {
    const float e = (l < len) ? __expf(att[l] - m) : 0.f;
    att[l] = e; ssum += e;
  }
  red[tid] = ssum; __syncthreads();
  for (int st = 128; st > 0; st >>= 1) {
    if (tid < st) red[tid] += red[tid + st];
    __syncthreads();
  }
  const float inv = 1.f / red[0];
  __syncthreads();
  for (int l = tid; l < L; l += 256) {
    const float p = att[l] * inv;
    att[l] = p;
    prob_out[(size_t)sb * L + l] = p;
  }
  __syncthreads();
  for (int h = tid; h < H; h += 256) {
    float acc = 0.f;
    for (int l = 0; l < len; ++l) acc += att[l] * eb[(size_t)l * H + h];
    ctx_out[(size_t)sb * H + h] = acc;
  }
}

// ---------------------------------------------------------------------------
// GRU gate combine (elementwise): gate order r, z, n (torch GRU).
// ---------------------------------------------------------------------------
__global__ void gru_combine(const float* __restrict__ gi, const float* __restrict__ gh,
                            const float* __restrict__ hprev, float* __restrict__ hnew,
                            int SB, int H) {
  const int idx = blockIdx.x * blockDim.x + threadIdx.x;
  if (idx >= SB * H) return;
  const int m = idx / H, j = idx % H;
  const float* gim = gi + (size_t)m * 3 * H;
  const float* ghm = gh + (size_t)m * 3 * H;
  const float ir = gim[j],      iz = gim[H + j],  in_ = gim[2 * H + j];
  const float hr = ghm[j],      hz = ghm[H + j],  hn  = ghm[2 * H + j];
  const float r = 1.f / (1.f + __expf(-(ir + hr)));
  const float z = 1.f / (1.f + __expf(-(iz + hz)));
  const float n = tanhf(in_ + r * hn);
  hnew[idx] = (1.f - z) * n + z * hprev[idx];
}

// ---------------------------------------------------------------------------
// Fused per-(s,b) finish: switch-gate sw, vocab softmax, pointer scatter-add
// (ds_add_f32 LDS atomics), output write, argmax, next-input embedding gather.
// Dynamic LDS: V floats (80 KB for V=20000 — fits CDNA5's 320 KB WGP LDS).
// ---------------------------------------------------------------------------
__global__ __launch_bounds__(256)
void final_out(const float* __restrict__ logits, const float* __restrict__ prob,
               const float* __restrict__ hnew, const float* __restrict__ ctxv,
               const float* __restrict__ xin,
               const float* __restrict__ w_ratio, const float* __restrict__ b_ratio,
               const int* __restrict__ story, const float* __restrict__ emb,
               float* __restrict__ out, float* __restrict__ xnext,
               int B, int L, int H, int V, int T, int t) {
  extern __shared__ float fin[];       // [V]
  __shared__ float red[256];
  __shared__ int   redi[256];
  __shared__ float s_sw;
  const int sb = blockIdx.x;
  const int b  = sb % B;
  const int tid = threadIdx.x;

  // sw = sigmoid(w_ratio . [h_new; ctx; x] + b_ratio)
  float acc = 0.f;
  for (int j = tid; j < 3 * H; j += 256) {
    float v;
    if (j < H)          v = hnew[(size_t)sb * H + j];
    else if (j < 2 * H) v = ctxv[(size_t)sb * H + (j - H)];
    else                v = xin[(size_t)sb * H + (j - 2 * H)];
    acc += w_ratio[j] * v;
  }
  red[tid] = acc; __syncthreads();
  for (int st = 128; st > 0; st >>= 1) { if (tid < st) red[tid] += red[tid + st]; __syncthreads(); }
  if (tid == 0) s_sw = 1.f / (1.f + __expf(-(red[0] + b_ratio[0])));
  __syncthreads();
  const float sw = s_sw;

  const float* lrow = logits + (size_t)sb * V;
  float m = -1e30f;
  for (int v = tid; v < V; v += 256) m = fmaxf(m, lrow[v]);
  red[tid] = m; __syncthreads();
  for (int st = 128; st > 0; st >>= 1) { if (tid < st) red[tid] = fmaxf(red[tid], red[tid + st]); __syncthreads(); }
  m = red[0]; __syncthreads();
  float ssum = 0.f;
  for (int v = tid; v < V; v += 256) { const float e = __expf(lrow[v] - m); fin[v] = e; ssum += e; }
  red[tid] = ssum; __syncthreads();
  for (int st = 128; st > 0; st >>= 1) { if (tid < st) red[tid] += red[tid + st]; __syncthreads(); }
  const float scale = sw / red[0];               // sw * softmax
  __syncthreads();
  for (int v = tid; v < V; v += 256) fin[v] *= scale;
  __syncthreads();
  const float onemsw = 1.f - sw;
  for (int l = tid; l < L; l += 256) {
    const float p = prob[(size_t)sb * L + l];
    if (p != 0.f) atomicAdd(&fin[story[b * L + l]], onemsw * p);
  }
  __syncthreads();
  float bm = -1e30f; int bi = 0;
  float* orow = out + ((size_t)sb * T + t) * V;
  for (int v = tid; v < V; v += 256) {
    const float f = fin[v];
    orow[v] = f;
    if (f > bm) { bm = f; bi = v; }
  }
  red[tid] = bm; redi[tid] = bi; __syncthreads();
  for (int st = 128; st > 0; st >>= 1) {
    if (tid < st) {
      if (red[tid + st] > red[tid] ||
          (red[tid + st] == red[tid] && redi[tid + st] < redi[tid])) {
        red[tid] = red[tid + st]; redi[tid] = redi[tid + st];
      }
    }
    __syncthreads();
  }
  const int pred = redi[0];
  for (int h = tid; h < H; h += 256)
    xnext[(size_t)sb * H + h] = emb[(size_t)pred * H + h];
}

// ---------------------------------------------------------------------------
extern "C" void kernel_launch(void* const* d_in, const int* in_sizes, int n_in,
                              void* d_out, int out_size, void* d_ws, size_t ws_size,
                              hipStream_t stream) {
  (void)n_in; (void)ws_size;
  const float* enc     = (const float*)d_in[1];
  const float* emb     = (const float*)d_in[2];
  const float* slot    = (const float*)d_in[3];
  const float* w_prior = (const float*)d_in[4];
  const float* b_prior = (const float*)d_in[5];
  const float* w_gate  = (const float*)d_in[6];
  const float* b_gate  = (const float*)d_in[7];
  const float* w_ratio = (const float*)d_in[8];
  const float* b_ratio = (const float*)d_in[9];
  const float* w_ih    = (const float*)d_in[10];
  const float* w_hh    = (const float*)d_in[11];
  const float* b_ih    = (const float*)d_in[12];
  const float* b_hh    = (const float*)d_in[13];
  const int*   story   = (const int*)d_in[14];
  const int*   lens    = (const int*)d_in[15];

  const int H = in_sizes[5];              // b_prior
  const int B = in_sizes[15];             // encoded_lens
  const int S = in_sizes[3] / H;          // slot_emb
  const int V = in_sizes[2] / H;          // emb_weight
  const int G = in_sizes[7];              // b_gate
  const int L = in_sizes[14] / B;         // story
  const int SB = S * B;
  const int T = (int)(((size_t)out_size - (size_t)SB * G) / ((size_t)SB * V));

  float* out       = (float*)d_out;
  float* gates_out = out + (size_t)SB * T * V;   // [S,B,G] after [S,B,T,V]

  // workspace carve (floats, 256B-aligned chunks)
  float* ws = (float*)d_ws;
  size_t off = 0;
  auto carve = [&](size_t n) { float* p = ws + off; off += (n + 63) & ~(size_t)63; return p; };
  float* q      = carve((size_t)S * H);
  float* probw  = carve((size_t)SB * L);
  float* ctxv   = carve((size_t)SB * H);
  float* gi     = carve((size_t)SB * 3 * H);
  float* gh     = carve((size_t)SB * 3 * H);
  float* logits = carve((size_t)SB * V);
  float* h0     = carve((size_t)SB * H);
  float* h1     = carve((size_t)SB * H);
  float* x0     = carve((size_t)SB * H);
  float* x1     = carve((size_t)SB * H);

  hipMemsetAsync(h0, 0, (size_t)SB * H * sizeof(float), stream);  // h_0 = zeros

  auto gemm = [&](const float* A, const float* W, const float* bias, float* C,
                  int M, int N, int K) {
    const int tilesM = (M + 15) / 16;
    const int tilesN = (N + 15) / 16;
    if ((M % 16 == 0) && (N % 16 == 0) && (K % 4 == 0) &&
        tilesM >= MT && tilesN >= NT) {
      const int mGroups = (tilesM + MT - 1) / MT;
      const int nGroups = (tilesN + NT - 1) / NT;
      const int waves   = mGroups * nGroups;
      const int blocks  = (waves + 3) / 4;        // 4 waves / block
      if (K == 400)
        gemm_fast<400><<<blocks, 128, 0, stream>>>(A, W, bias, C, M, N, K);
      else
        gemm_fast<0><<<blocks, 128, 0, stream>>>(A, W, bias, C, M, N, K);
    } else {
      const int waves  = tilesM * tilesN;
      const int blocks = (waves + 3) / 4;
      gemm_small<<<blocks, 128, 0, stream>>>(A, W, bias, C, M, N, K);
    }
  };

  const size_t attn_shmem = (size_t)(H + L) * sizeof(float);
  const size_t fin_shmem  = (size_t)V * sizeof(float);   // 80 KB << 320 KB WGP LDS

  // q = slot_emb @ w_prior^T + b_prior      [S,H]
  gemm(slot, w_prior, b_prior, q, S, H, H);
  // prior attention (query broadcast over b): ctx -> x0, prob scratch
  attn_softmax_ctx<<<SB, 256, attn_shmem, stream>>>(q, H, 0, enc, lens, probw, x0, B, L, H);
  // gates = ctx @ w_gate^T + b_gate         [S,B,G]
  gemm(x0, w_gate, b_gate, gates_out, SB, G, H);

  float* hc = h0; float* hn = h1; float* xc = x0; float* xn = x1;
  for (int t = 0; t < T; ++t) {
    gemm(xc, w_ih, b_ih, gi, SB, 3 * H, H);              // gi = x @ w_ih^T + b_ih
    gemm(hc, w_hh, b_hh, gh, SB, 3 * H, H);              // gh = h @ w_hh^T + b_hh
    const int tot = SB * H;
    gru_combine<<<(tot + 255) / 256, 256, 0, stream>>>(gi, gh, hc, hn, SB, H);
    // decoder attention with h_new
    attn_softmax_ctx<<<SB, 256, attn_shmem, stream>>>(hn, B * H, H, enc, lens,
                                                      probw, ctxv, B, L, H);
    // vocab logits = h_new @ emb^T  (dominant GEMM: [SB,H] x [H,V])
    gemm(hn, emb, nullptr, logits, SB, V, H);
    // fused softmax + pointer scatter + argmax + output + next-input gather
    final_out<<<SB, 256, fin_shmem, stream>>>(logits, probw, hn, ctxv, xc,
                                              w_ratio, b_ratio, story, emb,
                                              out, xn, B, L, H, V, T, t);
    float* tp = hc; hc = hn; hn = tp;
    tp = xc; xc = xn; xn = tp;
  }
}